// TriMipEncoding_28681791603373
// MI455X (gfx1250) — compile-verified
//
#include <hip/hip_runtime.h>

// ---------------------------------------------------------------------------
// TriMipEncoding for MI455X (gfx1250, wave32).
//   Stage 1: build mip pyramid levels 1..7 in d_ws.
//            Levels 1..5 use V_WMMA_F32_16X16X4_F32: the 2x2 box filter is
//            expressed as Out(16x16) = H(16x32, banded 0.25) x Rsum(32x16),
//            with Rsum = vertical pair-sum done in VALU. 8 chained K=4 WMMAs.
//   Stage 2: per (point, plane) thread: two bilinear quads (mip l0, l1),
//            float4 gathers, global_prefetch of the l1 texels, lerp, store.
// Pyramid (levels 1..7) per plane: 256^2+128^2+64^2+32^2+16^2+8^2+4^2 = 87376
// texels * 16 ch * 4 B * 3 planes = ~16.8 MB in d_ws.
// ---------------------------------------------------------------------------

typedef __attribute__((ext_vector_type(2))) float v2f;
typedef __attribute__((ext_vector_type(8))) float v8f;

#define PYR_TEXELS_PER_PLANE 87376LL
#define PYR_PLANE_F (PYR_TEXELS_PER_PLANE * 16LL)   // floats per plane in pyramid
#define FM_PLANE_F  (512LL * 512LL * 16LL)          // floats per plane in level 0

// per-plane texel offset of level l within the pyramid (levels 1..7; [0] unused)
__constant__ int c_off[8] = {0, 0, 65536, 81920, 86016, 87040, 87296, 87360};

// ---------------------------------------------------------------------------
// WMMA downsample: one wave computes a 16(x) x 16(ch) output tile for one
// (plane, y_out) row segment.  Wout must be a multiple of 16, input width 2*Wout.
// ---------------------------------------------------------------------------
__global__ void __launch_bounds__(256)
trimip_downsample_wmma(const float* __restrict__ src, long long srcPS,
                       float* __restrict__ dst, long long dstPS,
                       int Wout, int nTiles) {
    const int wave = threadIdx.x >> 5;
    const int lane = threadIdx.x & 31;
    const int tile = blockIdx.x * 8 + wave;
    if (tile >= nTiles) return;             // uniform per wave: EXEC stays all-1s

    const int tilesX = Wout >> 4;
    const int tpp    = tilesX * Wout;       // tiles per plane
    const int plane  = tile / tpp;
    int r            = tile - plane * tpp;
    const int yOut   = r / tilesX;
    const int x0     = (r - yOut * tilesX) << 4;
    const int Win    = Wout << 1;

    const float* s0 = src + (long long)plane * srcPS + (long long)(2 * yOut) * Win * 16;
    const float* s1 = s0 + (long long)Win * 16;

    const int n    = lane & 15;             // channel (N dim)
    const int half = lane >> 4;
    const int xin0 = x0 << 1;

    v8f acc = {};
#pragma unroll
    for (int kb = 0; kb < 8; ++kb) {
        // A layout (16x4 f32): lanes hold M = lane&15; VGPR0 <-> K = 4kb + 2*half,
        // VGPR1 <-> K = 4kb + 2*half + 1.  Banded H: nonzero iff M == K>>1.
        const float w = (n == (2 * kb + half)) ? 0.25f : 0.0f;
        v2f a; a.x = w; a.y = w;
        // B layout (4x16 f32): same K mapping; B[K][n] = Rsum[xin0+K][n]
        const int xA = xin0 + 4 * kb + 2 * half;
        v2f b;
        b.x = s0[(long long)xA * 16 + n]       + s1[(long long)xA * 16 + n];
        b.y = s0[(long long)(xA + 1) * 16 + n] + s1[(long long)(xA + 1) * 16 + n];
        acc = __builtin_amdgcn_wmma_f32_16x16x4_f32(
            /*neg_a=*/false, a, /*neg_b=*/false, b,
            /*c_mod=*/(short)0, acc, /*reuse_a=*/false, /*reuse_b=*/false);
    }

    // D layout: VGPR rr <-> M = rr + 8*half, N = lane&15
    float* dp = dst + (long long)plane * dstPS;
#pragma unroll
    for (int rr = 0; rr < 8; ++rr) {
        const int m = rr + half * 8;
        dp[((long long)yOut * Wout + x0 + m) * 16 + n] = acc[rr];
    }
}

// Scalar downsample for tiny levels (output width 8, 4).
__global__ void trimip_downsample_scalar(const float* __restrict__ src, long long srcPS,
                                         float* __restrict__ dst, long long dstPS,
                                         int Wout) {
    const int tid = blockIdx.x * blockDim.x + threadIdx.x;
    const int total = 3 * Wout * Wout * 16;
    if (tid >= total) return;
    const int c = tid & 15;
    int t = tid >> 4;
    const int xo = t % Wout; t /= Wout;
    const int yo = t % Wout;
    const int plane = t / Wout;
    const int Win = Wout << 1;
    const float* s = src + (long long)plane * srcPS;
    const long long b00 = ((long long)(2 * yo) * Win + 2 * xo) * 16 + c;
    const long long rowstep = (long long)Win * 16;
    const float v = 0.25f * (s[b00] + s[b00 + 16] + s[b00 + rowstep] + s[b00 + rowstep + 16]);
    dst[(long long)plane * dstPS + ((long long)yo * Wout + xo) * 16 + c] = v;
}

// ---------------------------------------------------------------------------
// Sampling kernel
// ---------------------------------------------------------------------------
struct TexQuad {
    const float* p00; const float* p01; const float* p10; const float* p11;
    float fx, fy;
};

__device__ __forceinline__ int clampi(int v, int lo, int hi) {
    return v < lo ? lo : (v > hi ? hi : v);
}

__device__ __forceinline__ TexQuad make_quad(const float* __restrict__ fm,
                                             const float* __restrict__ pyr,
                                             int plane, int l, float u, float v) {
    const int size = 512 >> l;
    const float* base = (l == 0)
        ? fm + (long long)plane * FM_PLANE_F
        : pyr + (long long)plane * PYR_PLANE_F + (long long)c_off[l] * 16;
    const float fs = (float)size;
    const float px = u * fs - 0.5f;
    const float py = v * fs - 0.5f;
    const float x0f = floorf(px);
    const float y0f = floorf(py);
    TexQuad q;
    q.fx = px - x0f;
    q.fy = py - y0f;
    const int x0 = (int)x0f, y0 = (int)y0f;
    const int xa = clampi(x0, 0, size - 1), xb = clampi(x0 + 1, 0, size - 1);
    const int ya = clampi(y0, 0, size - 1), yb = clampi(y0 + 1, 0, size - 1);
    q.p00 = base + ((long long)ya * size + xa) * 16;
    q.p01 = base + ((long long)ya * size + xb) * 16;
    q.p10 = base + ((long long)yb * size + xa) * 16;
    q.p11 = base + ((long long)yb * size + xb) * 16;
    return q;
}

__device__ __forceinline__ float4 lerp4(float4 a, float4 b, float t) {
    float4 r;
    r.x = a.x * (1.0f - t) + b.x * t;
    r.y = a.y * (1.0f - t) + b.y * t;
    r.z = a.z * (1.0f - t) + b.z * t;
    r.w = a.w * (1.0f - t) + b.w * t;
    return r;
}

__device__ __forceinline__ float4 bil4(const TexQuad& q, int cc) {
    const float4 f00 = *(const float4*)(q.p00 + cc * 4);
    const float4 f01 = *(const float4*)(q.p01 + cc * 4);
    const float4 f10 = *(const float4*)(q.p10 + cc * 4);
    const float4 f11 = *(const float4*)(q.p11 + cc * 4);
    const float4 top = lerp4(f00, f01, q.fx);
    const float4 bot = lerp4(f10, f11, q.fx);
    return lerp4(top, bot, q.fy);
}

__global__ void __launch_bounds__(256)
trimip_sample(const float* __restrict__ x, const float* __restrict__ level,
              const float* __restrict__ fm, const float* __restrict__ pyr,
              float* __restrict__ out, int N) {
    const int tid = blockIdx.x * blockDim.x + threadIdx.x;
    if (tid >= 3 * N) return;
    const int p  = tid / 3;
    const int pl = tid - p * 3;

    const float xv = x[p * 3 + 0];
    const float yv = x[p * 3 + 1];
    const float zv = x[p * 3 + 2];
    // plane0 -> (y,z), plane1 -> (x,z), plane2 -> (x,y)
    const float u = (pl == 0) ? yv : xv;
    const float v = (pl == 2) ? yv : zv;

    const float lv  = fminf(fmaxf(level[p], 0.0f), 7.0f);
    const float l0f = floorf(lv);
    const float f   = lv - l0f;
    const int l0 = (int)l0f;
    const int l1 = (l0 + 1 < 7) ? (l0 + 1) : 7;

    const TexQuad q0 = make_quad(fm, pyr, pl, l0, u, v);
    const TexQuad q1 = make_quad(fm, pyr, pl, l1, u, v);

    // Hide the dependent-gather latency of the second mip level:
    // emits global_prefetch_b8 (gfx1250 prefetch path).
    __builtin_prefetch(q1.p00, 0, 0);
    __builtin_prefetch(q1.p01, 0, 0);
    __builtin_prefetch(q1.p10, 0, 0);
    __builtin_prefetch(q1.p11, 0, 0);

    float* op = out + (long long)p * 48 + pl * 16;
#pragma unroll
    for (int cc = 0; cc < 4; ++cc) {
        const float4 s0 = bil4(q0, cc);
        const float4 s1 = bil4(q1, cc);
        *(float4*)(op + cc * 4) = lerp4(s0, s1, f);
    }
}

// ---------------------------------------------------------------------------
extern "C" void kernel_launch(void* const* d_in, const int* in_sizes, int n_in,
                              void* d_out, int out_size, void* d_ws, size_t ws_size,
                              hipStream_t stream) {
    const float* x     = (const float*)d_in[0];   // (N,3)
    const float* level = (const float*)d_in[1];   // (N,1)
    const float* fm    = (const float*)d_in[2];   // (3,512,512,16)
    // d_in[3] = n_levels (==8, hardcoded)
    const int N = in_sizes[0] / 3;

    float* pyr = (float*)d_ws;                    // levels 1..7, ~16.8 MB
    const long long PS = PYR_PLANE_F;
    const int off[8] = {0, 0, 65536, 81920, 86016, 87040, 87296, 87360};

    // level 0 -> 1 (Wout = 256)
    {
        const int Wout = 256;
        const int nT = 3 * (Wout >> 4) * Wout;
        trimip_downsample_wmma<<<(nT + 7) / 8, 256, 0, stream>>>(
            fm, FM_PLANE_F, pyr, PS, Wout, nT);
    }
    // levels 2..5 (Wout = 128, 64, 32, 16)
    for (int l = 2; l <= 5; ++l) {
        const int Wout = 512 >> l;
        const int nT = 3 * (Wout >> 4) * Wout;
        trimip_downsample_wmma<<<(nT + 7) / 8, 256, 0, stream>>>(
            pyr + (long long)off[l - 1] * 16, PS,
            pyr + (long long)off[l] * 16, PS, Wout, nT);
    }
    // levels 6..7 (Wout = 8, 4) — input width < 32, scalar path
    for (int l = 6; l <= 7; ++l) {
        const int Wout = 512 >> l;
        const int total = 3 * Wout * Wout * 16;
        trimip_downsample_scalar<<<(total + 255) / 256, 256, 0, stream>>>(
            pyr + (long long)off[l - 1] * 16, PS,
            pyr + (long long)off[l] * 16, PS, Wout);
    }

    // sampling: one thread per (point, plane)
    const long long threads = 3LL * N;
    trimip_sample<<<(int)((threads + 255) / 256), 256, 0, stream>>>(
        x, level, fm, pyr, (float*)d_out, N);
}